// ProteinGraphModel_56453050138695
// MI455X (gfx1250) — compile-verified
//
#include <hip/hip_runtime.h>
#include <hip/hip_bf16.h>

// ---------------------------------------------------------------------------
// MI455X (gfx1250, wave32) implementation of the protein-GNN forward pass.
// Memory-bound problem (~1.3 GB traffic @ 23.3 TB/s vs ~30 GFLOP) -> keep
// fp32 and use V_WMMA_F32_16X16X4_F32 for all dense layers so the ALU side
// never limits. Weights are staged TRANSPOSED in LDS so B-fragment pairs
// (K, K+1) are contiguous ds loads (lower to ds_load_2addr_b64).
// Dense layers run IN-PLACE on one LDS tile (the full GEMM lands in acc VGPRs
// before any store; per-wave LDS is in-order) -> 152 KB LDS, 2 blocks/WGP.
// All bandwidth-dominant global streams use float4 (b128) accesses.
// ---------------------------------------------------------------------------

typedef __attribute__((ext_vector_type(2))) float v2f;
typedef __attribute__((ext_vector_type(8))) float v8f;

#define NATOM 100000
#define KNBR  8
#define HDIM  128
#define NEDGE 800000
#define NPROT 1024
#define EPN   16384
#define ECN   131072
#define WPAD  132          // padded K stride for LDS weight / activation tiles
#define NWAVE 8            // waves per block in GEMM kernels
#define TPB   (NWAVE * 32)

// d_out layout (flat fp32 elements, tuple return order)
#define OUT_PRO   0
#define OUT_PRED  (OUT_PRO + NPROT * HDIM)            // 131072
#define OUT_FEAT  (OUT_PRED + NATOM)                  // 231072
#define OUT_MERGE (OUT_FEAT + NATOM * HDIM)           // 13031072
#define OUT_CROSS (OUT_MERGE + 2 * (EPN + 2 * ECN))   // 13588128
#define MERGE_M   (EPN + 2 * ECN)                     // 278528

// ---------------------------------------------------------------------------
// Wave-level 16xKin @ KinxH GEMM using V_WMMA_F32_16X16X4_F32.
// A fragment (16x4 fp32): lanes 0-15 hold K=0,1 / lanes 16-31 hold K=2,3 for
// row M=lane%16.  B fragment (4x16): same K striping, N=lane%16, so with the
// weight stored transposed (wT[n][k]) both fragment halves are contiguous.
// ---------------------------------------------------------------------------
__device__ inline void wave_gemm(const float* __restrict__ aT, int lda,
                                 const float* __restrict__ wT, int ldw,
                                 int kIter, v8f acc[8]) {
  const int lane = threadIdx.x & 31;
  const int half = lane >> 4;
  const int mr   = lane & 15;
  for (int kt = 0; kt < kIter; ++kt) {
    const int k0 = 4 * kt + 2 * half;
    v2f a;
    a.x = aT[mr * lda + k0];
    a.y = aT[mr * lda + k0 + 1];
#pragma unroll
    for (int nt = 0; nt < 8; ++nt) {
      v2f b;
      const float* wp = wT + (nt * 16 + mr) * ldw + k0;
      b.x = wp[0];
      b.y = wp[1];
      acc[nt] = __builtin_amdgcn_wmma_f32_16x16x4_f32(
          false, a, false, b, (short)0, acc[nt], false, false);
    }
  }
}

// Stage W (kin x HDIM, row-major) transposed into LDS: wsh[n*WPAD + k] = W[k][n]
// (global side is scattered but L2-resident; LDS side is bank-conflict-free)
__device__ inline void loadWT(const float* __restrict__ W, int kin,
                              float* __restrict__ wsh) {
  const int kpad = (kin + 3) & ~3;
  const int tot  = HDIM * kpad;
  for (int idx = threadIdx.x; idx < tot; idx += blockDim.x) {
    const int n = idx / kpad;
    const int k = idx - n * kpad;
    wsh[n * WPAD + k] = (k < kin) ? W[k * HDIM + n] : 0.0f;
  }
}

// One dense layer IN-PLACE on a 16-row tile: x = (relu?)(x @ W + bscale*bias)
__device__ inline void layer16(float* __restrict__ xio,
                               const float* __restrict__ wsh,
                               const float* __restrict__ bias, float bscale,
                               int kin, bool doRelu) {
  const int lane = threadIdx.x & 31;
  const int half = lane >> 4;
  const int mr   = lane & 15;
  v8f acc[8] = {};
  wave_gemm(xio, WPAD, wsh, WPAD, (kin + 3) >> 2, acc);
#pragma unroll
  for (int nt = 0; nt < 8; ++nt) {
    const float bv = bias[nt * 16 + mr] * bscale;
#pragma unroll
    for (int r = 0; r < 8; ++r) {
      float v = acc[nt][r] + bv;       // C layout: row = r + 8*half, col = nt*16+mr
      if (doRelu) v = fmaxf(v, 0.0f);
      xio[(r + 8 * half) * WPAD + nt * 16 + mr] = v;
    }
  }
}

// ---------------------------------------------------------------------------
// Kernel 1: atom pipeline  a1(+sum over K) -> a2 -> b1 -> b2 -> c1(+geom)
//           -> c2 -> scale  =>  feat0  (one 16-atom tile per wave)
// ---------------------------------------------------------------------------
__global__ __launch_bounds__(TPB) void node_pipeline(
    const float* __restrict__ emb, const int* __restrict__ atom_cat,
    const float* __restrict__ atom_dist, const float* __restrict__ xfeat,
    const float* __restrict__ a1W, const float* __restrict__ a1b,
    const float* __restrict__ a2W, const float* __restrict__ a2b,
    const float* __restrict__ b1W, const float* __restrict__ b1b,
    const float* __restrict__ b2W, const float* __restrict__ b2b,
    const float* __restrict__ c1W, const float* __restrict__ c1b,
    const float* __restrict__ c2W, const float* __restrict__ c2b,
    const float* __restrict__ sW, const float* __restrict__ sb,
    float* __restrict__ feat0) {
  __shared__ float wsh[HDIM * WPAD];        // transposed weights of current layer
  __shared__ float xbuf[NWAVE][16][WPAD];   // activation tile (in-place layers)
  __shared__ float atile[NWAVE][16][40];    // (atom,k) feature rows for layer a1

  const int w    = threadIdx.x >> 5;
  const int lane = threadIdx.x & 31;
  const int half = lane >> 4;
  const int mr   = lane & 15;
  const int a0   = blockIdx.x * (NWAVE * 16) + w * 16;   // first atom of this wave

  // ---- layer a1 (33 -> 128) on 8 m-tiles of 16 (atom,k) rows; ReLU then
  //      sum the 8 k-rows of each atom before the a2 GEMM ----
  loadWT(a1W, 33, wsh);
  __syncthreads();
  for (int mt = 0; mt < 8; ++mt) {
    // stage 16 rows: 32 emb floats as 8 float4 chunks + inv-dist column
    for (int j = lane; j < 16 * 8; j += 32) {
      const int r  = j >> 3;
      const int c4 = j & 7;
      const int la = 2 * mt + (r >> 3);
      const int atom = min(a0 + la, NATOM - 1);
      const int kk   = r & 7;
      const int v    = atom_cat[atom * KNBR + kk];
      *(float4*)&atile[w][r][c4 * 4] =
          *(const float4*)&emb[v * 32 + c4 * 4];
    }
    for (int j = lane; j < 16; j += 32) {   // lanes 0-15: cols 32..35
      const int la = 2 * mt + (j >> 3);
      const int atom = min(a0 + la, NATOM - 1);
      const int kk   = j & 7;
      float4 t;
      t.x = 1.0f / atom_dist[atom * KNBR + kk];
      t.y = 0.0f; t.z = 0.0f; t.w = 0.0f;
      *(float4*)&atile[w][j][32] = t;
    }
    __syncthreads();
    v8f acc[8] = {};
    wave_gemm(&atile[w][0][0], 40, wsh, WPAD, 9, acc);
#pragma unroll
    for (int nt = 0; nt < 8; ++nt) {
      const float bv = a1b[nt * 16 + mr];
      float s = 0.0f;
#pragma unroll
      for (int r = 0; r < 8; ++r) s += fmaxf(acc[nt][r] + bv, 0.0f);
      // rows 0-7 of the m-tile (lanes 0-15) = atom 2*mt; rows 8-15 = atom 2*mt+1
      xbuf[w][2 * mt + half][nt * 16 + mr] = s;
    }
    __syncthreads();
  }

  // ---- a2 (note: Sum_k relu_k @ a2W + 8*a2b) ----
  loadWT(a2W, HDIM, wsh);
  __syncthreads();
  layer16(&xbuf[w][0][0], wsh, a2b, 8.0f, HDIM, false);
  __syncthreads();
  loadWT(b1W, HDIM, wsh);
  __syncthreads();
  layer16(&xbuf[w][0][0], wsh, b1b, 1.0f, HDIM, true);
  __syncthreads();
  loadWT(b2W, HDIM, wsh);
  __syncthreads();
  layer16(&xbuf[w][0][0], wsh, b2b, 1.0f, HDIM, false);
  // append geom column (x[:,0]) for c1 (kin = 129, pad to 132)
  if (lane < 16) {
    const int atom = min(a0 + lane, NATOM - 1);
    float4 t;
    t.x = xfeat[atom * 4 + 0];
    t.y = 0.0f; t.z = 0.0f; t.w = 0.0f;
    *(float4*)&xbuf[w][lane][HDIM] = t;
  }
  __syncthreads();
  loadWT(c1W, 129, wsh);
  __syncthreads();
  layer16(&xbuf[w][0][0], wsh, c1b, 1.0f, 129, true);
  __syncthreads();
  loadWT(c2W, HDIM, wsh);
  __syncthreads();
  layer16(&xbuf[w][0][0], wsh, c2b, 1.0f, HDIM, false);
  __syncthreads();
  loadWT(sW, HDIM, wsh);
  __syncthreads();
  layer16(&xbuf[w][0][0], wsh, sb, 1.0f, HDIM, false);
  __syncthreads();

  // store feat0 (float4 / b128)
  for (int j = lane; j < 16 * 32; j += 32) {
    const int r  = j >> 5;
    const int c4 = j & 31;
    const int atom = a0 + r;
    if (atom < NATOM)
      *(float4*)&feat0[atom * HDIM + c4 * 4] = *(const float4*)&xbuf[w][r][c4 * 4];
  }
}

// ---------------------------------------------------------------------------
// Kernel 0: zero the segment-sum accumulator (b128 stores)
// ---------------------------------------------------------------------------
__global__ void zero_kernel(float4* __restrict__ p, int n4) {
  int i = blockIdx.x * blockDim.x + threadIdx.x;
  if (i < n4) p[i] = make_float4(0.f, 0.f, 0.f, 0.f);
}

// ---------------------------------------------------------------------------
// Kernel 2: edges  msg = relu(feat0[src] + [d, ea] @ edgeW + b); agg[dst]+=msg
// One wave per edge; b128 gather of feat0, 4x global_atomic_add_f32 per lane.
// ---------------------------------------------------------------------------
__global__ __launch_bounds__(256) void edge_kernel(
    const int* __restrict__ edge_index, const float* __restrict__ edge_attr,
    const float* __restrict__ pos, const float* __restrict__ eW,
    const float* __restrict__ eb, const float* __restrict__ feat0,
    float* __restrict__ agg) {
  const int eid  = blockIdx.x * 8 + (threadIdx.x >> 5);
  const int lane = threadIdx.x & 31;
  if (eid >= NEDGE) return;
  const int s = edge_index[eid];
  const int d = edge_index[NEDGE + eid];
  const float dx = pos[s * 3 + 0] - pos[d * 3 + 0];
  const float dy = pos[s * 3 + 1] - pos[d * 3 + 1];
  const float dz = pos[s * 3 + 2] - pos[d * 3 + 2];
  const float dist = sqrtf(dx * dx + dy * dy + dz * dz);
  const float ea = edge_attr[eid];
  const int c0 = lane * 4;
  const float4 f0 = *(const float4*)&feat0[s * HDIM + c0];
  const float4 w0 = *(const float4*)&eW[c0];
  const float4 w1 = *(const float4*)&eW[HDIM + c0];
  const float4 bb = *(const float4*)&eb[c0];
  float m[4];
  m[0] = fmaxf(f0.x + dist * w0.x + ea * w1.x + bb.x, 0.0f);
  m[1] = fmaxf(f0.y + dist * w0.y + ea * w1.y + bb.y, 0.0f);
  m[2] = fmaxf(f0.z + dist * w0.z + ea * w1.z + bb.z, 0.0f);
  m[3] = fmaxf(f0.w + dist * w0.w + ea * w1.w + bb.w, 0.0f);
#pragma unroll
  for (int j = 0; j < 4; ++j) unsafeAtomicAdd(&agg[d * HDIM + c0 + j], m[j]);
}

// ---------------------------------------------------------------------------
// Kernel 3: feat = mlp2(feat0+agg, g1, g2) + feat0 ; pred = sigmoid(feat@predW+b)
// ---------------------------------------------------------------------------
__global__ __launch_bounds__(TPB) void node_finish(
    const float* __restrict__ feat0, const float* __restrict__ agg,
    const float* __restrict__ g1W, const float* __restrict__ g1b,
    const float* __restrict__ g2W, const float* __restrict__ g2b,
    const float* __restrict__ predW, const float* __restrict__ predb,
    float* __restrict__ outFeat, float* __restrict__ outPred) {
  __shared__ float wsh[HDIM * WPAD];
  __shared__ float xbuf[NWAVE][16][WPAD];
  __shared__ float fbuf[NWAVE][16][HDIM];

  const int w    = threadIdx.x >> 5;
  const int lane = threadIdx.x & 31;
  const int a0   = blockIdx.x * (NWAVE * 16) + w * 16;

  loadWT(g1W, HDIM, wsh);
  for (int j = lane; j < 16 * 32; j += 32) {
    const int r  = j >> 5;
    const int c4 = j & 31;
    const int atom = min(a0 + r, NATOM - 1);
    const float4 f0 = *(const float4*)&feat0[atom * HDIM + c4 * 4];
    const float4 ag = *(const float4*)&agg[atom * HDIM + c4 * 4];
    *(float4*)&fbuf[w][r][c4 * 4] = f0;
    float4 z;
    z.x = f0.x + ag.x; z.y = f0.y + ag.y; z.z = f0.z + ag.z; z.w = f0.w + ag.w;
    *(float4*)&xbuf[w][r][c4 * 4] = z;
  }
  __syncthreads();
  layer16(&xbuf[w][0][0], wsh, g1b, 1.0f, HDIM, true);
  __syncthreads();
  loadWT(g2W, HDIM, wsh);
  __syncthreads();
  layer16(&xbuf[w][0][0], wsh, g2b, 1.0f, HDIM, false);
  __syncthreads();

  for (int j = lane; j < 16 * 32; j += 32) {
    const int r  = j >> 5;
    const int c4 = j & 31;
    float4 v = *(const float4*)&xbuf[w][r][c4 * 4];
    const float4 f0 = *(const float4*)&fbuf[w][r][c4 * 4];
    v.x += f0.x; v.y += f0.y; v.z += f0.z; v.w += f0.w;
    *(float4*)&xbuf[w][r][c4 * 4] = v;
    const int atom = a0 + r;
    if (atom < NATOM) *(float4*)&outFeat[atom * HDIM + c4 * 4] = v;
  }
  __syncthreads();
  if (lane < 16) {
    const int atom = a0 + lane;
    if (atom < NATOM) {
      const float4* xr = (const float4*)&xbuf[w][lane][0];
      const float4* pw = (const float4*)predW;
      float t = predb[0];
      for (int c4 = 0; c4 < 32; ++c4) {
        const float4 a = xr[c4], b = pw[c4];
        t += a.x * b.x + a.y * b.y + a.z * b.z + a.w * b.w;
      }
      outPred[atom] = 1.0f / (1.0f + __expf(-t));
    }
  }
}

// ---------------------------------------------------------------------------
// Kernel 4: pro_out = LayerNorm(pro_emb @ protW + protb) * ln_g + ln_b
// 1024x1280x128 WMMA GEMM, K chunked through LDS, fused LN.
// ---------------------------------------------------------------------------
__global__ __launch_bounds__(TPB) void protein_kernel(
    const float* __restrict__ pro_emb, const float* __restrict__ protW,
    const float* __restrict__ protb, const float* __restrict__ lng,
    const float* __restrict__ lnb, float* __restrict__ proOut) {
  __shared__ float wsh[HDIM * WPAD];
  __shared__ float xbuf[NWAVE][16][WPAD];

  const int w    = threadIdx.x >> 5;
  const int lane = threadIdx.x & 31;
  const int half = lane >> 4;
  const int mr   = lane & 15;
  const int r0   = blockIdx.x * (NWAVE * 16) + w * 16;

  v8f acc[8] = {};
  for (int ch = 0; ch < 10; ++ch) {          // 1280 = 10 * 128
    __syncthreads();
    for (int idx = threadIdx.x; idx < HDIM * HDIM; idx += TPB) {
      const int n = idx >> 7, k = idx & 127;
      wsh[n * WPAD + k] = protW[(ch * HDIM + k) * HDIM + n];
    }
    for (int j = lane; j < 16 * 32; j += 32) {
      const int r  = j >> 5;
      const int c4 = j & 31;
      *(float4*)&xbuf[w][r][c4 * 4] =
          *(const float4*)&pro_emb[(r0 + r) * 1280 + ch * HDIM + c4 * 4];
    }
    __syncthreads();
    wave_gemm(&xbuf[w][0][0], WPAD, wsh, WPAD, 32, acc);
  }
  __syncthreads();
#pragma unroll
  for (int nt = 0; nt < 8; ++nt) {
    const float bv = protb[nt * 16 + mr];
#pragma unroll
    for (int r = 0; r < 8; ++r)
      xbuf[w][r + 8 * half][nt * 16 + mr] = acc[nt][r] + bv;
  }
  __syncthreads();
  if (lane < 16) {
    const float4* xr = (const float4*)&xbuf[w][lane][0];
    float mu = 0.0f;
    for (int c4 = 0; c4 < 32; ++c4) {
      const float4 a = xr[c4];
      mu += a.x + a.y + a.z + a.w;
    }
    mu *= (1.0f / HDIM);
    float var = 0.0f;
    for (int c4 = 0; c4 < 32; ++c4) {
      const float4 a = xr[c4];
      var += (a.x - mu) * (a.x - mu) + (a.y - mu) * (a.y - mu) +
             (a.z - mu) * (a.z - mu) + (a.w - mu) * (a.w - mu);
    }
    var *= (1.0f / HDIM);
    const float rs  = rsqrtf(var + 1e-5f);
    const int   row = r0 + lane;
    for (int c = 0; c < HDIM; ++c)
      proOut[row * HDIM + c] = (xbuf[w][lane][c] - mu) * rs * lng[c] + lnb[c];
  }
}

// ---------------------------------------------------------------------------
// Kernel 5: cross_all = concat([1/prot_ind, 1/prot_dist, 1/prot_dist]) @ cW + cb
// float4 per thread (b128 stores on the 142 MB stream).
// ---------------------------------------------------------------------------
__global__ void cross_kernel(const float* __restrict__ prot_ind,
                             const float* __restrict__ prot_dist,
                             const float* __restrict__ cW,
                             const float* __restrict__ cb,
                             float4* __restrict__ out) {
  const int idx = blockIdx.x * blockDim.x + threadIdx.x;   // float4 index
  if (idx >= MERGE_M * (HDIM / 4)) return;
  const int r  = idx >> 5;          // 32 float4 per row
  const int c4 = idx & 31;
  float inv;
  if (r < EPN) {
    inv = 1.0f / prot_ind[r];
  } else {
    int rr = r - EPN;
    if (rr >= ECN) rr -= ECN;
    inv = 1.0f / prot_dist[rr];
  }
  const float4 wv = *(const float4*)&cW[c4 * 4];
  const float4 bv = *(const float4*)&cb[c4 * 4];
  float4 o;
  o.x = inv * wv.x + bv.x;
  o.y = inv * wv.y + bv.y;
  o.z = inv * wv.z + bv.z;
  o.w = inv * wv.w + bv.w;
  out[idx] = o;
}

// ---------------------------------------------------------------------------
// Kernel 6: merge = concat([pro_edge, [m0, m1+NPROT], [m1+NPROT, m0]], axis=1)
// (int32 bits written into the fp32 output buffer region)
// ---------------------------------------------------------------------------
__global__ void merge_kernel(const int* __restrict__ pro_edge,
                             const int* __restrict__ mv,   // (2, ECN)
                             int* __restrict__ out) {
  const int idx = blockIdx.x * blockDim.x + threadIdx.x;
  if (idx >= 2 * MERGE_M) return;
  const int row = idx / MERGE_M;
  const int j   = idx - row * MERGE_M;
  int val;
  if (j < EPN) {
    val = pro_edge[row * EPN + j];
  } else if (j < EPN + ECN) {
    const int t = j - EPN;
    val = (row == 0) ? mv[t] : (mv[ECN + t] + NPROT);
  } else {
    const int t = j - EPN - ECN;
    val = (row == 0) ? (mv[ECN + t] + NPROT) : mv[t];
  }
  out[idx] = val;
}

// ---------------------------------------------------------------------------
extern "C" void kernel_launch(void* const* d_in, const int* in_sizes, int n_in,
                              void* d_out, int out_size, void* d_ws,
                              size_t ws_size, hipStream_t stream) {
  (void)in_sizes; (void)n_in; (void)out_size; (void)ws_size;
  // inputs (setup_inputs dict order; params flattened in insertion order)
  const float* pos       = (const float*)d_in[0];
  const int*   atom_cat  = (const int*)d_in[1];
  const float* atom_dist = (const float*)d_in[2];
  const float* xfeat     = (const float*)d_in[3];
  const int*   edge_idx  = (const int*)d_in[4];
  const float* edge_attr = (const float*)d_in[5];
  const int*   mergev    = (const int*)d_in[6];
  const int*   pro_edge  = (const int*)d_in[7];
  const float* prot_dist = (const float*)d_in[8];
  const float* prot_ind  = (const float*)d_in[9];
  const float* pro_emb   = (const float*)d_in[10];
  const float* a1W = (const float*)d_in[11]; const float* a1b = (const float*)d_in[12];
  const float* a2W = (const float*)d_in[13]; const float* a2b = (const float*)d_in[14];
  const float* b1W = (const float*)d_in[15]; const float* b1b = (const float*)d_in[16];
  const float* b2W = (const float*)d_in[17]; const float* b2b = (const float*)d_in[18];
  const float* c1W = (const float*)d_in[19]; const float* c1b = (const float*)d_in[20];
  const float* c2W = (const float*)d_in[21]; const float* c2b = (const float*)d_in[22];
  const float* sW  = (const float*)d_in[23]; const float* sb  = (const float*)d_in[24];
  const float* eW  = (const float*)d_in[25]; const float* eb  = (const float*)d_in[26];
  const float* pW  = (const float*)d_in[27]; const float* pb  = (const float*)d_in[28];
  const float* cW  = (const float*)d_in[29]; const float* cb  = (const float*)d_in[30];
  const float* g1W = (const float*)d_in[31]; const float* g1b = (const float*)d_in[32];
  const float* g2W = (const float*)d_in[33]; const float* g2b = (const float*)d_in[34];
  const float* prW = (const float*)d_in[35]; const float* prb = (const float*)d_in[36];
  const float* emb = (const float*)d_in[37];
  const float* lng = (const float*)d_in[38];
  const float* lnb = (const float*)d_in[39];

  float* out   = (float*)d_out;
  float* agg   = (float*)d_ws;                  // N*H floats
  float* feat0 = agg + (size_t)NATOM * HDIM;    // N*H floats

  const int nodeBlocks = (NATOM + NWAVE * 16 - 1) / (NWAVE * 16);

  zero_kernel<<<(NATOM * HDIM / 4 + 255) / 256, 256, 0, stream>>>(
      (float4*)agg, NATOM * HDIM / 4);

  node_pipeline<<<nodeBlocks, TPB, 0, stream>>>(
      emb, atom_cat, atom_dist, xfeat, a1W, a1b, a2W, a2b, b1W, b1b, b2W, b2b,
      c1W, c1b, c2W, c2b, sW, sb, feat0);

  edge_kernel<<<(NEDGE + 7) / 8, 256, 0, stream>>>(edge_idx, edge_attr, pos,
                                                   eW, eb, feat0, agg);

  node_finish<<<nodeBlocks, TPB, 0, stream>>>(feat0, agg, g1W, g1b, g2W, g2b,
                                              prW, prb, out + OUT_FEAT,
                                              out + OUT_PRED);

  protein_kernel<<<NPROT / (NWAVE * 16), TPB, 0, stream>>>(
      pro_emb, pW, pb, lng, lnb, out + OUT_PRO);

  cross_kernel<<<(MERGE_M * (HDIM / 4) + 255) / 256, 256, 0, stream>>>(
      prot_ind, prot_dist, cW, cb, (float4*)(out + OUT_CROSS));

  merge_kernel<<<(2 * MERGE_M + 255) / 256, 256, 0, stream>>>(
      pro_edge, mergev, (int*)(out + OUT_MERGE));
}